// CERNNModel_40450001993765
// MI455X (gfx1250) — compile-verified
//
#include <hip/hip_runtime.h>
#include <hip/hip_bf16.h>

// ---------------- problem constants ----------------
#define T_STEPS 512
#define BSZ     256
#define NRNN    1024
#define NIN     53
#define NOUT    33
#define KTOT    1088          // 1024 (hh) + 64 (ih padded)
#define ALPHA_F 0.2f
#define ONE_MINUS_ALPHA 0.8f

// ---------------- persistent-kernel geometry ----------------
#define NBLK    128           // 8 batch-chunks (32 rows) x 16 n-chunks (64 cols)
#define THR     128           // 4 wave32 per WG
#define KPW     1092          // padded LDS row stride for W tile (1092%64==4 -> no bank conflicts)
#define HSP     132           // padded LDS row stride for staged h chunk (132%64==4)
#define KC      128           // K-chunk staged per double-buffer slot
// LDS: W tile 64*KPW + 2 staging buffers 32*HSP each
#define LDS_FLOATS (64 * KPW + 2 * 32 * HSP)
#define LDS_BYTES  (LDS_FLOATS * 4)          // 313,344 B of 320 KB WGP LDS

typedef float v2f __attribute__((ext_vector_type(2)));
typedef float v8f __attribute__((ext_vector_type(8)));
typedef int   v4i __attribute__((ext_vector_type(4)));
typedef __attribute__((address_space(1))) v4i* as1_v4i;   // global (__device__) int4*
typedef __attribute__((address_space(3))) v4i* as3_v4i;   // LDS (__shared__) int4*

// fp32 WMMA: D(16x16) += A(16x4) * B(4x16), all f32 (exact fp32 math)
__device__ __forceinline__ v8f wmma_f32(v2f a, v2f b, v8f c) {
  return __builtin_amdgcn_wmma_f32_16x16x4_f32(false, a, false, b, (short)0, c,
                                               false, false);
}

// CDNA5 async global->LDS copy (ASYNCcnt-tracked DMA), 16 B per lane.
__device__ __forceinline__ void cp_g2l_16(const float* g, float* l) {
#if __has_builtin(__builtin_amdgcn_global_load_async_to_lds_b128)
  __builtin_amdgcn_global_load_async_to_lds_b128((as1_v4i)(void*)g,
                                                 (as3_v4i)(void*)l, 0, 0);
#else
  *(float4*)l = *(const float4*)g;   // synchronous fallback
#endif
}

__device__ __forceinline__ void wait_async0() {
#if __has_builtin(__builtin_amdgcn_s_wait_asynccnt)
  __builtin_amdgcn_s_wait_asynccnt(0);
#else
  asm volatile("s_wait_asynccnt 0x0" ::: "memory");
#endif
}

// Grid-wide barrier for the persistent kernel: monotonic counter (zeroed by the
// prep kernel on every kernel_launch, so graph replay is safe).
__device__ __forceinline__ void gsync(unsigned* ctr, unsigned& calls) {
  __syncthreads();
  calls += 1u;
  if (threadIdx.x == 0) {
    __threadfence();
    atomicAdd(ctr, 1u);
    const unsigned target = calls * (unsigned)NBLK;
    while (atomicAdd(ctr, 0u) < target) {
      __builtin_amdgcn_s_sleep(2);
    }
    __threadfence();
  }
  __syncthreads();
}

// Stage a 32-row x (NC4*4)-col f32 block from global into padded LDS rows.
template <int NC4>  // float4s per row: 32 (KC=128 cols) or 16 (x chunk, 64 cols)
__device__ __forceinline__ void stage(const float* __restrict__ g, int gstride,
                                      float* l, int tid) {
  constexpr int PER = (32 * NC4) / THR;  // 8 or 4 float4s per thread
  #pragma unroll
  for (int i = 0; i < PER; ++i) {
    const int f  = tid + i * THR;
    const int r  = f / NC4;              // NC4 is a power of two
    const int c4 = f % NC4;
    cp_g2l_16(g + (size_t)r * gstride + c4 * 4, l + r * HSP + c4 * 4);
  }
}

// Inner GEMM chunk: A fragments from staged LDS, B fragments from W-tile LDS.
template <int COLS>
__device__ __forceinline__ void mm_chunk(const float* aB, const float* w0,
                                         const float* w1, v8f& c0, v8f& c1) {
  #pragma unroll 8
  for (int kk = 0; kk < COLS; kk += 4) {
    const v2f a  = *(const v2f*)(aB + kk);
    const v2f b0 = *(const v2f*)(w0 + kk);
    const v2f b1 = *(const v2f*)(w1 + kk);
    c0 = wmma_f32(a, b0, c0);
    c1 = wmma_f32(a, b1, c1);
  }
}

// ---------------------------------------------------------------------------
// Prep 1: W_big[1024][1088] = [ w_hh_eff | w_ih_eff | 0 ], zero aux state.
// ---------------------------------------------------------------------------
__global__ void prep_weights(const float* __restrict__ wih,
                             const float* __restrict__ whh,
                             const float* __restrict__ ms,
                             const float* __restrict__ mv,
                             const float* __restrict__ mt,
                             const float* __restrict__ intra,
                             float* __restrict__ Wbig,
                             float* __restrict__ aux3k,
                             int*   __restrict__ flags,
                             unsigned* __restrict__ ctr) {
  const int n = blockIdx.x;        // row
  const int t = threadIdx.x;       // 256 threads
  const float* hrow = whh   + (size_t)n * NRNN;
  const float* irow = intra + (size_t)n * NRNN;
  float* dst = Wbig + (size_t)n * KTOT;
  for (int k = t; k < NRNN; k += 256) {
    const float w  = hrow[k];
    const float nz = (fabsf(w) > 0.01f) ? 1.0f : 0.0f;
    dst[k] = w * irow[k] * nz;
  }
  for (int j = t; j < 64; j += 256) {
    float v = 0.0f;
    if (j < NIN) {
      const size_t o = (size_t)n * NIN + j;
      v = wih[o] * (ms[o] + mv[o] + mt[o]);
    }
    dst[NRNN + j] = v;
  }
  if (n == 0) {
    for (int i = t; i < 3 * NRNN; i += 256) aux3k[i] = 0.0f;
    if (t < 128) flags[t] = 0;
    if (t == 0) *ctr = 0u;
  }
}

// ---------------------------------------------------------------------------
// Prep 2: x_pad[T*B][64] = x[T*B][53] zero-padded.
// ---------------------------------------------------------------------------
__global__ void prep_xpad(const float* __restrict__ x, float* __restrict__ xpad) {
  const int row = blockIdx.x * blockDim.x + threadIdx.x;
  if (row >= T_STEPS * BSZ) return;
  const float* src = x    + (size_t)row * NIN;
  float*       dst = xpad + (size_t)row * 64;
  #pragma unroll
  for (int j = 0; j < NIN; ++j) dst[j] = src[j];
  #pragma unroll
  for (int j = NIN; j < 64; ++j) dst[j] = 0.0f;
}

// ---------------------------------------------------------------------------
// Persistent RNN kernel: warm-up fixed point + 512-step scan.
// ---------------------------------------------------------------------------
__launch_bounds__(THR, 1)
__global__ void rnn_scan(const float* __restrict__ x,
                         const float* __restrict__ bias,
                         const float* __restrict__ Wbig,
                         const float* __restrict__ xpad,
                         float* __restrict__ h0v,
                         float* __restrict__ h1v,
                         float* __restrict__ h2v,
                         float* __restrict__ h_init,
                         int*   __restrict__ flags,
                         unsigned* __restrict__ ctr,
                         float* __restrict__ hidden) {
  extern __shared__ float ldsW[];              // [64][KPW] + 2x[32][HSP]
  float* hS0 = ldsW + 64 * KPW;
  float* hS1 = hS0 + 32 * HSP;
  const int wg      = blockIdx.x;
  const int b_chunk = wg & 7;
  const int n_chunk = wg >> 3;
  const int tid  = threadIdx.x;
  const int lane = tid & 31;
  const int wave = tid >> 5;

  // ---- load this WG's weight tile into padded LDS once ----
  {
    const float* src = Wbig + (size_t)n_chunk * 64 * KTOT;
    for (int i = tid; i < 64 * (KTOT / 4); i += THR) {
      const int r  = i / (KTOT / 4);
      const int c4 = i % (KTOT / 4);
      *(float4*)(ldsW + r * KPW + c4 * 4) =
          *(const float4*)(src + (size_t)r * KTOT + c4 * 4);
    }
  }
  __syncthreads();

  unsigned calls = 0;

  // =============== phase 1: fixed-point warm-up (1024-vector) =============
  const bool owner = (b_chunk == 0);
  int  stable = 0;
  bool done   = false;
  const float* xr0 = x;                        // x[0,0,:]
  const float* xr1 = x + (size_t)BSZ * NIN;    // x[1,0,:]

  for (int iter = 0; iter < 100; ++iter) {
    if (owner && tid < 64) {                   // h1 = step(h0, x0)
      const int n = n_chunk * 64 + tid;
      const float* wrow = ldsW + tid * KPW;
      float acc = 0.0f;
      for (int k = 0; k < NRNN; ++k) acc += h0v[k] * wrow[k];
      for (int j = 0; j < NIN;  ++j) acc += xr0[j] * wrow[NRNN + j];
      const float pre = acc + bias[n];
      h1v[n] = ONE_MINUS_ALPHA * h0v[n] + ALPHA_F * fmaxf(pre, 0.0f);
    }
    gsync(ctr, calls);
    if (owner && tid < 64) {                   // h2 = step(h1, x1) + tol check
      const int n = n_chunk * 64 + tid;
      const float* wrow = ldsW + tid * KPW;
      float acc = 0.0f;
      for (int k = 0; k < NRNN; ++k) acc += h1v[k] * wrow[k];
      for (int j = 0; j < NIN;  ++j) acc += xr1[j] * wrow[NRNN + j];
      const float pre = acc + bias[n];
      const float h2 = ONE_MINUS_ALPHA * h1v[n] + ALPHA_F * fmaxf(pre, 0.0f);
      h2v[n] = h2;
      const float h0 = h0v[n];
      if (fabsf(h2 - h0) > 0.1f + 1e-5f * fabsf(h0)) atomicOr(&flags[iter], 1);
    }
    gsync(ctr, calls);
    if (owner) {                               // reference carry logic, exact
      const bool close      = (flags[iter] == 0);
      const int  stable_new = close ? (stable + 1) : 0;
      const bool done_new   = done || (stable_new >= 4);
      if (tid < 64 && !done_new) {
        const int n = n_chunk * 64 + tid;
        h0v[n] = h2v[n];
      }
      stable = done ? stable : stable_new;
      done   = done_new;
    }
    gsync(ctr, calls);
  }

  // broadcast h0 vector to the full (256x1024) initial state
  for (int i = tid; i < 32 * 64; i += THR) {
    const int b = b_chunk * 32 + (i >> 6);
    const int n = n_chunk * 64 + (i & 63);
    h_init[(size_t)b * NRNN + n] = h0v[n];
  }
  gsync(ctr, calls);

  // =============== phase 2: 512-step scan, fp32 WMMA, async-staged A ======
  const int b_sub  = wave & 1;
  const int n_pair = wave >> 1;
  const int row    = lane & 15;
  const int koff   = (lane >> 4) << 1;         // lanes 16..31: K+2,K+3
  const int bRow0  = b_chunk * 32;             // WG's first batch row
  const int bBase  = bRow0 + b_sub * 16;
  const int nBase0 = n_chunk * 64 + n_pair * 32;
  const int nBase1 = nBase0 + 16;
  const float* w0row = ldsW + (n_pair * 32 +      row) * KPW + koff;
  const float* w1row = ldsW + (n_pair * 32 + 16 + row) * KPW + koff;
  const int aOff = (b_sub * 16 + row) * HSP + koff;

  for (int t = 0; t < T_STEPS; ++t) {
    const float* h_src = (t == 0) ? h_init
                                  : (hidden + (size_t)(t - 1) * BSZ * NRNN);
    const float* xbase = xpad + ((size_t)t * BSZ + bRow0) * 64;

    // prologue: stage K-chunk 0
    stage<KC / 4>(h_src + (size_t)bRow0 * NRNN, NRNN, hS0, tid);
    wait_async0();
    __syncthreads();

    v8f c0 = {}; v8f c1 = {};
    // 8 h-chunks of 128 + 1 x-chunk of 64; double-buffered async staging
    for (int c = 0; c < 9; ++c) {
      float* cur = (c & 1) ? hS1 : hS0;
      float* nxt = (c & 1) ? hS0 : hS1;
      if (c < 7) {
        stage<KC / 4>(h_src + (size_t)bRow0 * NRNN + (c + 1) * KC, NRNN, nxt, tid);
      } else if (c == 7) {
        stage<16>(xbase, 64, nxt, tid);
      }
      if (c < 8) mm_chunk<KC>(cur + aOff, w0row + c * KC, w1row + c * KC, c0, c1);
      else       mm_chunk<64>(cur + aOff, w0row + NRNN,   w1row + NRNN,   c0, c1);
      wait_async0();
      __syncthreads();
    }

    // epilogue: D vgpr r -> (M = r + 8*(lane>=16), N = lane&15)
    float* hout = hidden + (size_t)t * BSZ * NRNN;
    const int Mbase = (lane >> 4) * 8;
    const int ncol  = lane & 15;
    #pragma unroll
    for (int r = 0; r < 8; ++r) {
      const int b = bBase + Mbase + r;
      {
        const int n = nBase0 + ncol;
        const float pre = c0[r] + bias[n];
        const float hp  = h_src[(size_t)b * NRNN + n];
        hout[(size_t)b * NRNN + n] =
            ONE_MINUS_ALPHA * hp + ALPHA_F * fmaxf(pre, 0.0f);
      }
      {
        const int n = nBase1 + ncol;
        const float pre = c1[r] + bias[n];
        const float hp  = h_src[(size_t)b * NRNN + n];
        hout[(size_t)b * NRNN + n] =
            ONE_MINUS_ALPHA * hp + ALPHA_F * fmaxf(pre, 0.0f);
      }
    }
    gsync(ctr, calls);                         // hidden[t] complete everywhere
  }
}

// ---------------------------------------------------------------------------
// Output projection: out[t,b,o] = sum_m hidden[t,b,128+m] * w_out[o,m]
// ---------------------------------------------------------------------------
__global__ void out_proj(const float* __restrict__ hidden,
                         const float* __restrict__ w_out,
                         float* __restrict__ out) {
  const int idx = blockIdx.x * blockDim.x + threadIdx.x;
  if (idx >= T_STEPS * BSZ * NOUT) return;
  const int o  = idx % NOUT;
  const int rw = idx / NOUT;
  const float* hrow = hidden + (size_t)rw * NRNN + 128;
  const float* wrow = w_out + o * 64;
  float acc = 0.0f;
  #pragma unroll 8
  for (int m = 0; m < 64; ++m) acc += hrow[m] * wrow[m];
  out[idx] = acc;
}

// ---------------------------------------------------------------------------
extern "C" void kernel_launch(void* const* d_in, const int* in_sizes, int n_in,
                              void* d_out, int out_size, void* d_ws, size_t ws_size,
                              hipStream_t stream) {
  const float* x     = (const float*)d_in[0];
  const float* wih   = (const float*)d_in[1];
  const float* whh   = (const float*)d_in[2];
  const float* bias  = (const float*)d_in[3];
  const float* wout  = (const float*)d_in[4];
  const float* ms    = (const float*)d_in[5];
  const float* mv    = (const float*)d_in[6];
  const float* mt    = (const float*)d_in[7];
  const float* intra = (const float*)d_in[8];

  // workspace layout (~39.1 MB)
  float* Wbig   = (float*)d_ws;                          // 1024*1088
  float* xpad   = Wbig  + (size_t)NRNN * KTOT;           // 512*256*64
  float* h_init = xpad  + (size_t)T_STEPS * BSZ * 64;    // 256*1024
  float* h0v    = h_init + (size_t)BSZ * NRNN;           // 1024
  float* h1v    = h0v + NRNN;
  float* h2v    = h1v + NRNN;
  int*   flags  = (int*)(h2v + NRNN);                    // 128
  unsigned* ctr = (unsigned*)(flags + 128);              // 1

  // d_out = [ output (512*256*33) | hidden (512*256*1024) ]
  float* out    = (float*)d_out;
  float* hidden = out + (size_t)T_STEPS * BSZ * NOUT;

  prep_weights<<<NRNN, 256, 0, stream>>>(wih, whh, ms, mv, mt, intra,
                                         Wbig, h0v, flags, ctr);
  prep_xpad<<<(T_STEPS * BSZ + 255) / 256, 256, 0, stream>>>(x, xpad);

  rnn_scan<<<NBLK, THR, LDS_BYTES, stream>>>(x, bias, Wbig, xpad,
                                             h0v, h1v, h2v, h_init,
                                             flags, ctr, hidden);

  const int nout_elems = T_STEPS * BSZ * NOUT;
  out_proj<<<(nout_elems + 255) / 256, 256, 0, stream>>>(hidden, wout, out);
}